// vector_quantizer_67138928771109
// MI455X (gfx1250) — compile-verified
//
#include <hip/hip_runtime.h>

// Vector quantizer (VQ-VAE nearest-embed), N=500000, D=128, K=128, fp32.
// Both reference outputs equal W[argmin_k ||z - W_k||^2]; we compute the
// argmin with exact-fp32 tensor ops (v_wmma_f32_16x16x4_f32) and write the
// gathered code row to both output halves.

typedef __attribute__((ext_vector_type(2))) float v2f;
typedef __attribute__((ext_vector_type(4))) float v4f;
typedef __attribute__((ext_vector_type(8))) float v8f;

#define DIM 128
#define NCODE 128
#define WAVES_PER_BLOCK 8

__global__ __launch_bounds__(256) void vq_wmma_kernel(
    const float* __restrict__ z,   // [N, 128]
    const float* __restrict__ W,   // [128, 128]
    float* __restrict__ zq,        // [N, 128]
    float* __restrict__ emb,       // [N, 128]
    int nTiles)                    // N / 16
{
    // B fragments of W^T, pre-swizzled for V_WMMA_F32_16X16X4_F32:
    //   Bf[n][c][lane] = (W[n*16 + (lane&15)][4c + 2*(lane>>4)],
    //                     W[n*16 + (lane&15)][4c + 2*(lane>>4) + 1])
    // lanes 0-15 hold K-rows {0,1}, lanes 16-31 hold K-rows {2,3} of the 4x16 B tile.
    __shared__ v2f Bf[8][32][32];      // 64 KB
    __shared__ float wsq[NCODE];       // ||W_k||^2

    const int tid = threadIdx.x;

    // ---- cooperative LDS fill (all 256 threads) ----
    for (int f = tid; f < 8 * 32 * 32; f += 256) {
        int n   = f >> 10;          // / 1024
        int rem = f & 1023;
        int c   = rem >> 5;
        int L   = rem & 31;
        int k   = n * 16 + (L & 15);
        int d   = 4 * c + 2 * (L >> 4);
        Bf[n][c][L] = *(const v2f*)(W + (size_t)k * DIM + d);
    }
    if (tid < NCODE) {
        const float* wr = W + (size_t)tid * DIM;
        float s = 0.f;
        #pragma unroll 8
        for (int d = 0; d < DIM; ++d) s += wr[d] * wr[d];
        wsq[tid] = s;
    }
    __syncthreads();

    const int wave   = tid >> 5;
    const int lane   = tid & 31;
    const int laneLo = lane & 15;
    const int hi     = lane >> 4;

    const int tile = blockIdx.x * WAVES_PER_BLOCK + wave;
    if (tile >= nTiles) return;      // wave-uniform; no further block syncs

    // ---- load the 16x128 z-tile as A fragments (16x4 layout, 32 K-chunks) ----
    // lane<16 holds row=lane, d = {4c, 4c+1}; lane>=16 holds row=lane-16, d = {4c+2, 4c+3}
    const float* zbase = z + (size_t)(tile * 16 + laneLo) * DIM + 2 * hi;
    v2f a[32];
    #pragma unroll
    for (int c = 0; c < 32; ++c) a[c] = *(const v2f*)(zbase + 4 * c);

    // Running per-(row-slot, code-lane) best of  D = z.W_k - 0.5*||W_k||^2
    // (argmin distance == argmax D). C-layout: VGPR j <-> row M = j + 8*hi,
    // code N = (lane&15) + 16*n.
    float bestVal[8];
    int   bestIdx[8];
    #pragma unroll
    for (int j = 0; j < 8; ++j) { bestVal[j] = -3.402823466e38f; bestIdx[j] = 0; }

    #pragma unroll
    for (int n = 0; n < 8; ++n) {
        const int kk = n * 16 + laneLo;
        const float ci = -0.5f * wsq[kk];
        v8f acc = {ci, ci, ci, ci, ci, ci, ci, ci};
        #pragma unroll
        for (int c = 0; c < 32; ++c) {
            acc = __builtin_amdgcn_wmma_f32_16x16x4_f32(
                /*neg_a=*/false, a[c],
                /*neg_b=*/false, Bf[n][c][lane],
                /*c_mod=*/(short)0, acc,
                /*reuse_a=*/false, /*reuse_b=*/false);
        }
        #pragma unroll
        for (int j = 0; j < 8; ++j) {
            float v = acc[j];
            if (v > bestVal[j] || (v == bestVal[j] && kk < bestIdx[j])) {
                bestVal[j] = v;
                bestIdx[j] = kk;
            }
        }
    }

    // ---- argmax across the 16 code-lanes of each half-wave (ties -> lower idx) ----
    #pragma unroll
    for (int j = 0; j < 8; ++j) {
        #pragma unroll
        for (int m = 1; m < 16; m <<= 1) {
            float ov = __shfl_xor(bestVal[j], m, 32);
            int   oi = __shfl_xor(bestIdx[j], m, 32);
            if (ov > bestVal[j] || (ov == bestVal[j] && oi < bestIdx[j])) {
                bestVal[j] = ov;
                bestIdx[j] = oi;
            }
        }
    }

    // ---- gather winning code rows and write both outputs (coalesced float4) ----
    const size_t rowBase = (size_t)tile * 16;
    #pragma unroll
    for (int j = 0; j < 8; ++j) {
        const int idxLo = __shfl(bestIdx[j], 0, 32);   // row j      (lanes 0-15 group)
        const int idxHi = __shfl(bestIdx[j], 16, 32);  // row j + 8  (lanes 16-31 group)
        {
            const size_t r = rowBase + j;
            v4f w = *(const v4f*)(W + (size_t)idxLo * DIM + lane * 4);
            *(v4f*)(zq  + r * DIM + lane * 4) = w;
            *(v4f*)(emb + r * DIM + lane * 4) = w;
        }
        {
            const size_t r = rowBase + j + 8;
            v4f w = *(const v4f*)(W + (size_t)idxHi * DIM + lane * 4);
            *(v4f*)(zq  + r * DIM + lane * 4) = w;
            *(v4f*)(emb + r * DIM + lane * 4) = w;
        }
    }
}

extern "C" void kernel_launch(void* const* d_in, const int* in_sizes, int n_in,
                              void* d_out, int out_size, void* d_ws, size_t ws_size,
                              hipStream_t stream) {
    (void)n_in; (void)d_ws; (void)ws_size; (void)out_size;

    const float* z = (const float*)d_in[0];   // [N, 128] fp32
    const float* W = (const float*)d_in[1];   // [128, 128] fp32
    float* out = (float*)d_out;               // [z_q (N*128) | emb (N*128)]

    const int N = in_sizes[0] / DIM;          // 500000 (multiple of 16)
    float* zq  = out;
    float* emb = out + (size_t)N * DIM;

    const int nTiles = N / 16;                // exact for N = 500000
    const int blocks = (nTiles + WAVES_PER_BLOCK - 1) / WAVES_PER_BLOCK;

    vq_wmma_kernel<<<blocks, 256, 0, stream>>>(z, W, zq, emb, nTiles);
}